// YoloXLoss_32195074851446
// MI455X (gfx1250) — compile-verified
//
#include <hip/hip_runtime.h>
#include <hip/hip_bf16.h>
#include <math.h>

// YOLOX loss (SimOTA assignment + IoU/obj/cls/L1 losses) for MI455X (gfx1250).
// B=32, A=8400 anchors, G=40 GTs, C=80 classes. Outputs: 6 f32 scalars.
//
// Pipeline (all on `stream`, deterministic, no float atomics):
//   kA: per-anchor Sum_c log1p(-sqrt(cls*obj)) via V_WMMA_F32_16X16X4_F32
//       (16 anchors per wave, B = ones -> matrix pipe as exact f32 adder tree),
//       plus fg_anchor (center-radius any-GT) flags.
//   kB: per (b,g) row: top-10 smallest (cost,idx) pairs + top-10 largest IoUs
//       -> dynamic_k and rank-k threshold pair (stable-argsort semantics).
//   kC: per anchor: recompute cost/iou (shared __noinline__ fn => bitwise match
//       with kB), matching / multi / best_gt resolution, loss contributions,
//       block tree-reduction to partials.
//   kD: single-thread deterministic final reduction -> 6 outputs.

#define NBATCH 32
#define NA     8400
#define NG     40
#define NC     80
#define ROWD   85
#define BAtot  (NBATCH * NA)          // 268800 = 128*2100 = 256*1050
#define NBLK_C (BAtot / 256)          // 1050

typedef float v2f __attribute__((ext_vector_type(2)));
typedef float v8f __attribute__((ext_vector_type(8)));

// ---------------------------------------------------------------------------
// Shared cost/iou evaluation. __noinline__ so kernels kB and kC call the SAME
// emitted code -> bitwise-identical results, making the exact equality test
// against the stored rank-k threshold valid.
// ---------------------------------------------------------------------------
struct CI { float cost; float iou; };

__device__ __attribute__((noinline)) CI cost_iou(
    const float* __restrict__ row, float S, int fg_a,
    float gx, float gy, float gw, float gh, int cls,
    float xs, float ys, float st)
{
    float px = row[0], py = row[1], pw = row[2], ph = row[3], obj = row[4];
    // pairwise IoU (cxcywh), en = strict overlap, zeroed for non-fg anchors
    float tlx = fmaxf(gx - gw * 0.5f, px - pw * 0.5f);
    float tly = fmaxf(gy - gh * 0.5f, py - ph * 0.5f);
    float brx = fminf(gx + gw * 0.5f, px + pw * 0.5f);
    float bry = fminf(gy + gh * 0.5f, py + ph * 0.5f);
    float ai  = ((tlx < brx) && (tly < bry)) ? (brx - tlx) * (bry - tly) : 0.0f;
    float iou = ai / (gw * gh + pw * ph - ai);
    if (!fg_a) iou = 0.0f;
    // center-radius in-box test
    float xc = (xs + 0.5f) * st, yc = (ys + 0.5f) * st, r = 1.5f * st;
    float mn = fminf(fminf(xc - (gx - r), gx + r - xc),
                     fminf(yc - (gy - r), gy + r - yc));
    // cls cost: -(S + logit(p_gc)), p = sqrt(cls_p * obj_p)
    float p    = sqrtf(row[5 + cls] * obj);
    float lgt  = logf(p) - log1pf(-p);
    float cost = (-S - lgt) + 3.0f * (-logf(iou + 1e-8f))
               + ((mn > 0.0f) ? 0.0f : 1000000.0f);
    if (!fg_a) cost = 1000000000.0f;
    CI o; o.cost = cost; o.iou = iou; return o;
}

// ---------------------------------------------------------------------------
// Kernel A: S[b,a] = Sum_c log1p(-sqrt(cls*obj)) via WMMA f32 16x16x4,
//           fg_anchor[b,a] = any_g(center-radius hit).
// One wave = 16 anchors. A-frag layout (16x4 f32): lanes 0-15 hold K=0,1;
// lanes 16-31 hold K=2,3; row M = lane&15. B = ones => D[m,n] = Sum_k A[m,k].
// D layout: lane 0 holds col N=0 of rows 0-7, lane 16 rows 8-15.
// ---------------------------------------------------------------------------
__global__ __launch_bounds__(256) void yolox_kA(
    const float* __restrict__ outputs, const float* __restrict__ labels,
    const float* __restrict__ xsArr, const float* __restrict__ ysArr,
    const float* __restrict__ stArr,
    float* __restrict__ Sbuf, int* __restrict__ fgA)
{
    int lane = threadIdx.x & 31;
    int wave = threadIdx.x >> 5;
    int m    = lane & 15;
    int base = blockIdx.x * 128 + wave * 16;   // 16 anchors per wave
    int aIdx = base + m;                       // grid exact: no bounds check,
                                               // EXEC all ones for WMMA
    const float* row = outputs + (size_t)aIdx * ROWD;
    float obj = row[4];
    int   kh  = (lane < 16) ? 0 : 2;

    v8f acc = {};
    v2f ones; ones.x = 1.0f; ones.y = 1.0f;
    #pragma unroll
    for (int ch = 0; ch < 20; ++ch) {
        int   c0 = ch * 4 + kh;
        float p0 = sqrtf(row[5 + c0] * obj);
        float p1 = sqrtf(row[6 + c0] * obj);
        v2f afrag; afrag.x = log1pf(-p0); afrag.y = log1pf(-p1);
        acc = __builtin_amdgcn_wmma_f32_16x16x4_f32(
            false, afrag, false, ones, (short)0, acc, false, false);
    }
    if (lane == 0 || lane == 16) {
        int o = base + ((lane == 16) ? 8 : 0);
        #pragma unroll
        for (int r2 = 0; r2 < 8; ++r2) Sbuf[o + r2] = acc[r2];
    }

    if (lane < 16) {
        int b = aIdx / NA;
        int a = aIdx % NA;
        float st = stArr[a], xs = xsArr[a], ys = ysArr[a];
        float xc = (xs + 0.5f) * st, yc = (ys + 0.5f) * st, r = 1.5f * st;
        const float* lb = labels + (size_t)b * NG * 5;
        int fg = 0;
        for (int g = 0; g < NG; ++g) {
            float gx = lb[g * 5 + 1], gy = lb[g * 5 + 2];
            float mn = fminf(fminf(xc - (gx - r), gx + r - xc),
                             fminf(yc - (gy - r), gy + r - yc));
            fg |= (mn > 0.0f) ? 1 : 0;
        }
        fgA[aIdx] = fg;
    }
}

// ---------------------------------------------------------------------------
// Kernel B: one block per (b,g) row. Per-thread sorted top-10 lists over its
// strided anchors, then deterministic 10-round LDS merge (k-way merge of
// sorted heads). Emits dynamic_k threshold pair (cost,idx) for rank<k test.
// ---------------------------------------------------------------------------
__global__ __launch_bounds__(256) void yolox_kB(
    const float* __restrict__ outputs, const float* __restrict__ labels,
    const float* __restrict__ xsArr, const float* __restrict__ ysArr,
    const float* __restrict__ stArr,
    const float* __restrict__ Sbuf, const int* __restrict__ fgA,
    float* __restrict__ thrC, int* __restrict__ thrI)
{
    int bg = blockIdx.x;
    int b  = bg / NG;
    int t  = threadIdx.x;
    const float* lb = labels + (size_t)bg * 5;
    int   cls = (int)lb[0];
    float gx = lb[1], gy = lb[2], gw = lb[3], gh = lb[4];

    float c10[10]; int i10[10]; float v10[10];
    #pragma unroll
    for (int j = 0; j < 10; ++j) { c10[j] = 3.4e38f; i10[j] = 0x7fffffff; v10[j] = -1.0f; }

    for (int a = t; a < NA; a += 256) {
        int aIdx = b * NA + a;
        const float* row = outputs + (size_t)aIdx * ROWD;
        CI ci = cost_iou(row, Sbuf[aIdx], fgA[aIdx],
                         gx, gy, gw, gh, cls, xsArr[a], ysArr[a], stArr[a]);
        float cost = ci.cost, iou = ci.iou;
        // insert into ascending (cost, idx) top-10
        if (cost < c10[9] || (cost == c10[9] && a < i10[9])) {
            int j = 9;
            while (j > 0 && (cost < c10[j-1] || (cost == c10[j-1] && a < i10[j-1]))) {
                c10[j] = c10[j-1]; i10[j] = i10[j-1]; --j;
            }
            c10[j] = cost; i10[j] = a;
        }
        // insert into descending iou top-10
        if (iou > v10[9]) {
            int j = 9;
            while (j > 0 && iou > v10[j-1]) { v10[j] = v10[j-1]; --j; }
            v10[j] = iou;
        }
    }

    __shared__ float sC[2560]; __shared__ int sI[2560]; __shared__ float sV[2560];
    __shared__ float rC[256];  __shared__ int rA[256];  __shared__ int rT[256];
    __shared__ float selC[10]; __shared__ int selI[10]; __shared__ float sumIou;
    #pragma unroll
    for (int j = 0; j < 10; ++j) {
        sC[t*10+j] = c10[j]; sI[t*10+j] = i10[j]; sV[t*10+j] = v10[j];
    }
    __syncthreads();

    // ---- top-10 largest IoU (sum in descending order, like top_k().sum())
    int ptr = 0;
    float isum = 0.0f;
    for (int rd = 0; rd < 10; ++rd) {
        float hv = (ptr < 10) ? sV[t*10 + ptr] : -2.0f;
        rC[t] = hv; rT[t] = t;
        __syncthreads();
        for (int s2 = 128; s2 > 0; s2 >>= 1) {
            if (t < s2) {
                if (rC[t+s2] > rC[t]) { rC[t] = rC[t+s2]; rT[t] = rT[t+s2]; }
            }
            __syncthreads();
        }
        if (t == rT[0]) ptr++;
        if (t == 0) isum += rC[0];
        __syncthreads();
    }
    if (t == 0) sumIou = isum;

    // ---- top-10 smallest (cost, idx) pairs
    ptr = 0;
    __syncthreads();
    for (int rd = 0; rd < 10; ++rd) {
        float hc; int hi;
        if (ptr < 10) { hc = sC[t*10 + ptr]; hi = sI[t*10 + ptr]; }
        else          { hc = 3.4e38f;        hi = 0x7fffffff;     }
        rC[t] = hc; rA[t] = hi; rT[t] = t;
        __syncthreads();
        for (int s2 = 128; s2 > 0; s2 >>= 1) {
            if (t < s2) {
                float oc = rC[t+s2]; int oa = rA[t+s2];
                if (oc < rC[t] || (oc == rC[t] && oa < rA[t])) {
                    rC[t] = oc; rA[t] = oa; rT[t] = rT[t+s2];
                }
            }
            __syncthreads();
        }
        if (t == rT[0]) ptr++;
        if (t == 0) { selC[rd] = rC[0]; selI[rd] = rA[0]; }
        __syncthreads();
    }
    if (t == 0) {
        int k = (int)sumIou;              // trunc toward zero, like astype(int32)
        if (k < 1)  k = 1;
        if (k > 10) k = 10;
        thrC[bg] = selC[k-1];
        thrI[bg] = selI[k-1];
    }
}

// ---------------------------------------------------------------------------
// Kernel C: per anchor, resolve assignment across 40 GTs + loss contributions.
// Deterministic LDS tree reduction -> per-block partials (5 sums).
// ---------------------------------------------------------------------------
__global__ __launch_bounds__(256) void yolox_kC(
    const float* __restrict__ outputs, const float* __restrict__ origin,
    const float* __restrict__ labels,
    const float* __restrict__ xsArr, const float* __restrict__ ysArr,
    const float* __restrict__ stArr,
    const float* __restrict__ Sbuf, const int* __restrict__ fgA,
    const float* __restrict__ thrC, const int* __restrict__ thrI,
    float* __restrict__ partials)
{
    int t    = threadIdx.x;
    int aIdx = blockIdx.x * 256 + t;     // exact grid, no bounds check
    int b    = aIdx / NA;
    int a    = aIdx % NA;
    const float* row = outputs + (size_t)aIdx * ROWD;
    float px = row[0], py = row[1], pw = row[2], ph = row[3], obj = row[4];
    float st = stArr[a], xs = xsArr[a], ys = ysArr[a];
    float area_p = pw * ph;
    float S = Sbuf[aIdx];
    int fg_a = fgA[aIdx];

    int   count = 0, firstG = -1, bestG = 0;
    float predSum = 0.0f, bestCost = 3.5e38f, bestIou = 0.0f;
    for (int g = 0; g < NG; ++g) {
        const float* lb = labels + (size_t)(b * NG + g) * 5;
        int   cls = (int)lb[0];
        float gx = lb[1], gy = lb[2], gw = lb[3], gh = lb[4];
        CI ci = cost_iou(row, S, fg_a, gx, gy, gw, gh, cls, xs, ys, st);
        float cost = ci.cost, iou = ci.iou;
        float tc = thrC[b * NG + g];
        int   ti = thrI[b * NG + g];
        bool  m0 = (cost < tc) || (cost == tc && a <= ti);   // rank < dynamic_k
        if (m0) { count++; if (firstG < 0) firstG = g; predSum += iou; }
        if (cost < bestCost) { bestCost = cost; bestG = g; bestIou = iou; }
    }

    bool  fg; int mg; float predIou;
    if (count > 1)       { fg = true;  mg = bestG;  predIou = bestIou; }   // multi -> onehot(best)
    else if (count == 1) { fg = true;  mg = firstG; predIou = predSum; }
    else                 { fg = false; mg = 0;      predIou = 0.0f;    }

    // obj BCE over ALL anchors
    float tt   = fg ? 1.0f : 0.0f;
    float lobj = fmaxf(obj, 0.0f) - obj * tt + log1pf(expf(-fabsf(obj)));

    float liou = 0.0f, l1 = 0.0f, lcls = 0.0f;
    if (fg) {
        const float* lm = labels + (size_t)(b * NG + mg) * 5;
        int   mcls = (int)lm[0];
        float rx = lm[1], ry = lm[2], rw = lm[3], rh = lm[4];
        // iou_elem (with +1e-16 in denominator)
        float tlx = fmaxf(rx - rw * 0.5f, px - pw * 0.5f);
        float tly = fmaxf(ry - rh * 0.5f, py - ph * 0.5f);
        float brx = fminf(rx + rw * 0.5f, px + pw * 0.5f);
        float bry = fminf(ry + rh * 0.5f, py + ph * 0.5f);
        float ai  = ((tlx < brx) && (tly < bry)) ? (brx - tlx) * (bry - tly) : 0.0f;
        float ie  = ai / (area_p + rw * rh - ai + 1e-16f);
        liou = 1.0f - ie * ie;
        // L1
        float t0 = rx / st - xs, t1 = ry / st - ys;
        float t2 = logf(rw / st + 1e-8f), t3 = logf(rh / st + 1e-8f);
        const float* op = origin + (size_t)aIdx * 4;
        l1 = fabsf(op[0]-t0) + fabsf(op[1]-t1) + fabsf(op[2]-t2) + fabsf(op[3]-t3);
        // cls BCE vs onehot(mcls)*pred_iou
        for (int c = 0; c < NC; ++c) {
            float x  = row[5 + c];
            float tc2 = (c == mcls) ? predIou : 0.0f;
            lcls += fmaxf(x, 0.0f) - x * tc2 + log1pf(expf(-fabsf(x)));
        }
    }

    __shared__ float red[256];
    float vals[5] = { fg ? 1.0f : 0.0f, liou, lobj, lcls, l1 };
    for (int j = 0; j < 5; ++j) {
        red[t] = vals[j];
        __syncthreads();
        for (int s2 = 128; s2 > 0; s2 >>= 1) {
            if (t < s2) red[t] += red[t + s2];
            __syncthreads();
        }
        if (t == 0) partials[blockIdx.x * 5 + j] = red[0];
        __syncthreads();
    }
}

// ---------------------------------------------------------------------------
// Kernel D: deterministic final reduction + 6 outputs.
// ---------------------------------------------------------------------------
__global__ void yolox_kD(const float* __restrict__ partials, float* __restrict__ out)
{
    float s0 = 0, s1 = 0, s2 = 0, s3 = 0, s4 = 0;
    for (int i = 0; i < NBLK_C; ++i) {
        s0 += partials[i*5+0]; s1 += partials[i*5+1]; s2 += partials[i*5+2];
        s3 += partials[i*5+3]; s4 += partials[i*5+4];
    }
    float numfg = fmaxf(s0, 1.0f);
    float li = 5.0f * s1 / numfg;
    float lo = s2 / numfg;
    float lc = s3 / numfg;
    float ll = s4 / numfg;
    out[0] = li + lo + lc + ll;
    out[1] = li;
    out[2] = lo;
    out[3] = lc;
    out[4] = ll;
    out[5] = numfg / (float)(NBATCH * NG);
}

extern "C" void kernel_launch(void* const* d_in, const int* in_sizes, int n_in,
                              void* d_out, int out_size, void* d_ws, size_t ws_size,
                              hipStream_t stream)
{
    const float* outputs = (const float*)d_in[0];
    const float* origin  = (const float*)d_in[1];
    const float* labels  = (const float*)d_in[2];
    const float* xsArr   = (const float*)d_in[3];
    const float* ysArr   = (const float*)d_in[4];
    const float* stArr   = (const float*)d_in[5];
    float* out = (float*)d_out;

    // workspace layout (~2.2 MB)
    float* Sbuf     = (float*)d_ws;                    // BAtot floats
    int*   fgA      = (int*)(Sbuf + BAtot);            // BAtot ints
    float* thrC     = (float*)(fgA + BAtot);           // B*G floats
    int*   thrI     = (int*)(thrC + NBATCH * NG);      // B*G ints
    float* partials = (float*)(thrI + NBATCH * NG);    // NBLK_C*5 floats

    yolox_kA<<<BAtot / 128, 256, 0, stream>>>(outputs, labels, xsArr, ysArr, stArr,
                                              Sbuf, fgA);
    yolox_kB<<<NBATCH * NG, 256, 0, stream>>>(outputs, labels, xsArr, ysArr, stArr,
                                              Sbuf, fgA, thrC, thrI);
    yolox_kC<<<BAtot / 256, 256, 0, stream>>>(outputs, origin, labels, xsArr, ysArr,
                                              stArr, Sbuf, fgA, thrC, thrI, partials);
    yolox_kD<<<1, 1, 0, stream>>>(partials, out);
}